// TimeTransformerBlock_5660766896214
// MI455X (gfx1250) — compile-verified
//
#include <hip/hip_runtime.h>

typedef __attribute__((ext_vector_type(2))) float v2f;
typedef __attribute__((ext_vector_type(8))) float v8f;

__device__ __forceinline__ v8f wmma4(v2f a, v2f b, v8f c) {
  // D = A(16x4) * B(4x16) + C(16x16), fp32 WMMA
  return __builtin_amdgcn_wmma_f32_16x16x4_f32(false, a, false, b, (short)0, c,
                                               false, false);
}

// ---------------- pack: x[B=2,C=128,T=512,F=64] -> seq[N=128,T,C] ----------
__global__ __launch_bounds__(256) void k_pack(const float* __restrict__ x,
                                              float* __restrict__ seq) {
  int t = blockIdx.x & 511;
  int b = blockIdx.x >> 9;
  __shared__ float tile[128 * 65];
  for (int idx = threadIdx.x; idx < 8192; idx += 256) {
    int f = idx & 63, c = idx >> 6;  // f fastest -> coalesced on x
    tile[c * 65 + f] = x[(((long)(b * 128 + c)) * 512 + t) * 64 + f];
  }
  __syncthreads();
  for (int idx = threadIdx.x; idx < 8192; idx += 256) {
    int c = idx & 127, f = idx >> 7;  // c fastest -> coalesced on seq
    seq[((long)(b * 64 + f) * 512 + t) * 128 + c] = tile[c * 65 + f];
  }
}

// ---------------- layernorm over C=128, one wave per row -------------------
__global__ __launch_bounds__(256) void k_ln_rows(const float* __restrict__ X,
                                                 float* __restrict__ Y,
                                                 const float* __restrict__ g,
                                                 const float* __restrict__ b,
                                                 long rows) {
  int lane = threadIdx.x & 31;
  long row = (long)blockIdx.x * 8 + (threadIdx.x >> 5);
  if (row >= rows) return;
  const float4* xr = (const float4*)(X + row * 128);
  float4 v = xr[lane];
  float s = v.x + v.y + v.z + v.w;
#pragma unroll
  for (int m = 16; m >= 1; m >>= 1) s += __shfl_xor(s, m, 32);
  float mean = s * (1.f / 128.f);
  float dx = v.x - mean, dy = v.y - mean, dz = v.z - mean, dw = v.w - mean;
  float q = dx * dx + dy * dy + dz * dz + dw * dw;
#pragma unroll
  for (int m = 16; m >= 1; m >>= 1) q += __shfl_xor(q, m, 32);
  float rstd = rsqrtf(q * (1.f / 128.f) + 1e-5f);
  int c = lane * 4;
  float4 o;
  o.x = dx * rstd * g[c + 0] + b[c + 0];
  o.y = dy * rstd * g[c + 1] + b[c + 1];
  o.z = dz * rstd * g[c + 2] + b[c + 2];
  o.w = dw * rstd * g[c + 3] + b[c + 3];
  ((float4*)(Y + row * 128))[lane] = o;
}

// ---------------- generic WMMA fp32 GEMM: O[m,n] = sum_k A[m,k]*W[n,k]+bias[n]
// one 16x16 output tile per wave; K stepped by 4 (v_wmma_f32_16x16x4_f32).
__global__ __launch_bounds__(256) void k_gemm(const float* __restrict__ A, int lda,
                                              long aGrpOff,
                                              const float* __restrict__ W, int ldw,
                                              const float* __restrict__ bias,
                                              float* __restrict__ O, int ldo,
                                              long oGrpOff, int M, int Nout, int K) {
  A += (long)blockIdx.z * aGrpOff;
  O += (long)blockIdx.z * oGrpOff;
  int lane = threadIdx.x & 31;
  int ntiles = Nout >> 4;
  long tile = (long)blockIdx.x * 8 + (threadIdx.x >> 5);
  long mt = tile / ntiles;
  int nt = (int)(tile % ntiles);
  if (mt >= (M >> 4)) return;
  int m0 = (int)mt << 4, n0 = nt << 4;
  int half = lane >> 4, l16 = lane & 15, koff = half * 2;
  const float* arow = A + (long)(m0 + l16) * lda + koff;  // A frag: M=l16, K=koff..
  const float* wrow = W + (long)(n0 + l16) * ldw + koff;  // B frag: N=l16 (B=W^T)
  v8f acc = {};
  for (int k = 0; k < K; k += 4) {
    v2f a = *(const v2f*)(arow + k);
    v2f b = *(const v2f*)(wrow + k);
    acc = wmma4(a, b, acc);
  }
  float bn = bias[n0 + l16];
  int mbase = m0 + half * 8;
#pragma unroll
  for (int i = 0; i < 8; ++i)
    O[(long)(mbase + i) * ldo + n0 + l16] = acc[i] + bn;
}

// ---------------- grouped GRU scan: one wave per (n,g); whh in registers ----
__global__ __launch_bounds__(32) void k_gru_scan(const float* __restrict__ xp, // [N,T,4,96]
                                                 const float* __restrict__ whh, // [96,32]
                                                 const float* __restrict__ bhh, // [96]
                                                 float* __restrict__ seq) {     // [N,T,128] +=
  int lane = threadIdx.x;  // d index 0..31
  int n = blockIdx.x >> 2, g = blockIdx.x & 3;
  float wr[32], wz[32], wn[32];
#pragma unroll
  for (int d = 0; d < 32; ++d) {
    wr[d] = whh[lane * 32 + d];
    wz[d] = whh[(32 + lane) * 32 + d];
    wn[d] = whh[(64 + lane) * 32 + d];
  }
  float br = bhh[lane], bz = bhh[32 + lane], bn = bhh[64 + lane];
  float h = 0.f;
  const float* xpb = xp + (long)n * 512 * 384 + g * 96;
  float* sb = seq + (long)n * 512 * 128 + g * 32;
  for (int t = 0; t < 512; ++t) {
    const float* x3 = xpb + (long)t * 384;
    float xr = x3[lane], xz = x3[32 + lane], xn = x3[64 + lane];
    float gr = br, gz = bz, gn = bn;
#pragma unroll
    for (int d = 0; d < 32; ++d) {
      float hd = __shfl(h, d, 32);  // lowers to v_readlane (constant lane)
      gr += wr[d] * hd;
      gz += wz[d] * hd;
      gn += wn[d] * hd;
    }
    float r = 1.f / (1.f + __expf(-(xr + gr)));
    float z = 1.f / (1.f + __expf(-(xz + gz)));
    float nn = tanhf(xn + r * gn);
    h = (1.f - z) * nn + z * h;
    float* so = sb + (long)t * 128;
    so[lane] = so[lane] + h;  // residual add in place
  }
}

// ---------------- windowed flash attention, one 16-row q-tile per wave ------
// qkv: [N,T,384] (Q|K|V each 128 = 4 heads x 32). out: [N,T,128].
__global__ __launch_bounds__(256) void k_attn(const float* __restrict__ qkv,
                                              float* __restrict__ out) {
  __shared__ float Pl[8][16 * 18];
  int lane = threadIdx.x & 31;
  int wv = threadIdx.x >> 5;
  long wid = (long)blockIdx.x * 8 + wv;
  int qt = wid & 31;
  int h = (wid >> 5) & 3;
  int n = (int)(wid >> 7);
  int q0 = qt << 4;
  int half = lane >> 4, l16 = lane & 15, koff = half * 2;
  const float* base = qkv + (long)n * 512 * 384;
  const float* Qb = base + h * 32;
  const float* Kb = base + 128 + h * 32;
  const float* Vb = base + 256 + h * 32;
  v2f qf[8];
  {
    const float* qrow = Qb + (long)(q0 + l16) * 384 + koff;
#pragma unroll
    for (int kk = 0; kk < 8; ++kk) qf[kk] = *(const v2f*)(qrow + kk * 4);
  }
  v8f O0 = {}, O1 = {};
  float rmax[8], rsum[8];
#pragma unroll
  for (int i = 0; i < 8; ++i) { rmax[i] = -1e30f; rsum[i] = 0.f; }
  int kt_lo = (q0 > 100) ? ((q0 - 100) >> 4) : 0;
  float* P = Pl[wv];
  for (int kt = kt_lo; kt <= qt; ++kt) {
    int tk0 = kt << 4;
    v8f S = {};
    const float* krow = Kb + (long)(tk0 + l16) * 384 + koff;
#pragma unroll
    for (int kk = 0; kk < 8; ++kk) {
      v2f bf = *(const v2f*)(krow + kk * 4);
      S = wmma4(qf[kk], bf, S);  // S[q, k] over Dh=32
    }
    int kpos = tk0 + l16;
#pragma unroll
    for (int i = 0; i < 8; ++i) {
      int qpos = q0 + half * 8 + i;
      int diff = qpos - kpos;
      bool ok = (diff >= 0) && (diff <= 100);
      float s = S[i] * 0.17677669529663687f;  // 1/sqrt(32)
      float sm = ok ? s : -1e30f;
      float m = sm;
#pragma unroll
      for (int xm = 8; xm >= 1; xm >>= 1) m = fmaxf(m, __shfl_xor(m, xm, 16));
      float nmax = fmaxf(rmax[i], m);
      float sc = __expf(rmax[i] - nmax);
      float p = ok ? __expf(s - nmax) : 0.f;
      float ps = p;
#pragma unroll
      for (int xm = 8; xm >= 1; xm >>= 1) ps += __shfl_xor(ps, xm, 16);
      rsum[i] = rsum[i] * sc + ps;
      rmax[i] = nmax;
      O0[i] *= sc;
      O1[i] *= sc;
      P[(half * 8 + i) * 18 + l16] = p;  // C-layout -> LDS
    }
    asm volatile("s_wait_dscnt 0x0" ::: "memory");  // per-wave; DS is in-order
#pragma unroll
    for (int kk2 = 0; kk2 < 4; ++kk2) {  // K2 = 16
      int kb = kk2 * 4 + koff;
      v2f a2;
      a2.x = P[l16 * 18 + kb];
      a2.y = P[l16 * 18 + kb + 1];
      const float* v0 = Vb + (long)(tk0 + kb) * 384;
      v2f b0, b1;
      b0.x = v0[l16];       b0.y = v0[384 + l16];
      b1.x = v0[16 + l16];  b1.y = v0[384 + 16 + l16];
      O0 = wmma4(a2, b0, O0);
      O1 = wmma4(a2, b1, O1);
    }
    asm volatile("" ::: "memory");
  }
  float* ob = out + (long)n * 512 * 128 + h * 32;
#pragma unroll
  for (int i = 0; i < 8; ++i) {
    float rinv = 1.f / rsum[i];
    int m = q0 + half * 8 + i;
    ob[(long)m * 128 + l16] = O0[i] * rinv;
    ob[(long)m * 128 + 16 + l16] = O1[i] * rinv;
  }
}

// ---------------- unpack: out[b,c,t,f] = seq[n,t,c] + proj[n,t,c] -----------
__global__ __launch_bounds__(256) void k_unpack(const float* __restrict__ seq,
                                                const float* __restrict__ proj,
                                                float* __restrict__ out) {
  int t = blockIdx.x & 511;
  int b = blockIdx.x >> 9;
  __shared__ float tile[128 * 65];
  for (int idx = threadIdx.x; idx < 8192; idx += 256) {
    int c = idx & 127, f = idx >> 7;
    long off = ((long)(b * 64 + f) * 512 + t) * 128 + c;
    tile[c * 65 + f] = seq[off] + proj[off];
  }
  __syncthreads();
  for (int idx = threadIdx.x; idx < 8192; idx += 256) {
    int f = idx & 63, c = idx >> 6;
    out[(((long)(b * 128 + c)) * 512 + t) * 64 + f] = tile[c * 65 + f];
  }
}

extern "C" void kernel_launch(void* const* d_in, const int* in_sizes, int n_in,
                              void* d_out, int out_size, void* d_ws, size_t ws_size,
                              hipStream_t stream) {
  const float* x = (const float*)d_in[0];
  const float* ln1_g = (const float*)d_in[1];
  const float* ln1_b = (const float*)d_in[2];
  const float* gru_wih = (const float*)d_in[3];   // [96,32]
  const float* gru_whh = (const float*)d_in[4];   // [96,32]
  const float* gru_bih = (const float*)d_in[5];
  const float* gru_bhh = (const float*)d_in[6];
  const float* ln2_g = (const float*)d_in[7];
  const float* ln2_b = (const float*)d_in[8];
  const float* in_proj_w = (const float*)d_in[9];   // [384,128]
  const float* in_proj_b = (const float*)d_in[10];
  const float* out_proj_w = (const float*)d_in[11]; // [128,128]
  const float* out_proj_b = (const float*)d_in[12];

  const long S = (long)128 * 512 * 128;  // 8.39M floats
  float* seq = (float*)d_ws;             // [N,T,C]
  float* lnb = seq + S;                  // LN output, later attention output
  float* big = seq + 2 * S;              // xp [N,T,384], later qkv, later proj out

  // seq = permute(x); lnb = LN1(seq)
  k_pack<<<1024, 256, 0, stream>>>(x, seq);
  k_ln_rows<<<8192, 256, 0, stream>>>(seq, lnb, ln1_g, ln1_b, 65536L);
  // xp = lnb_group @ wih^T + bih   (M=65536, N=96, K=32, grid.z = 4 groups)
  k_gemm<<<dim3(3072, 1, 4), 256, 0, stream>>>(lnb, 128, 32, gru_wih, 32, gru_bih,
                                               big, 384, 96, 65536, 96, 32);
  // seq += GRU(xp)  (512 independent scans, one wave each)
  k_gru_scan<<<512, 32, 0, stream>>>(big, gru_whh, gru_bhh, seq);
  // lnb = LN2(seq)
  k_ln_rows<<<8192, 256, 0, stream>>>(seq, lnb, ln2_g, ln2_b, 65536L);
  // qkv = lnb @ in_proj_w^T + b   (M=65536, N=384, K=128)
  k_gemm<<<dim3(12288, 1, 1), 256, 0, stream>>>(lnb, 128, 0, in_proj_w, 128,
                                                in_proj_b, big, 384, 0, 65536, 384, 128);
  // lnb = attention(qkv)  (windowed flash, 16384 wave tiles)
  k_attn<<<2048, 256, 0, stream>>>(big, lnb);
  // big = lnb @ out_proj_w^T + b  (M=65536, N=128, K=128)
  k_gemm<<<dim3(4096, 1, 1), 256, 0, stream>>>(lnb, 128, 0, out_proj_w, 128,
                                               out_proj_b, big, 128, 0, 65536, 128, 128);
  // out = permute(seq + big)
  k_unpack<<<1024, 256, 0, stream>>>(seq, big, (float*)d_out);
}